// Dwt2d_7928509629347
// MI455X (gfx1250) — compile-verified
//
#include <hip/hip_runtime.h>
#include <hip/hip_bf16.h>
#include <stdint.h>

// ---- vector typedefs -------------------------------------------------------
typedef float v4f __attribute__((ext_vector_type(4)));
typedef unsigned int uint32x4 __attribute__((ext_vector_type(4)));
typedef int int32x8 __attribute__((ext_vector_type(8)));
typedef int int32x4 __attribute__((ext_vector_type(4)));

// Tile geometry: each block DMAs 32 input rows (of 256 floats) into LDS via
// the Tensor Data Mover, producing 16 output rows x 128 cols x 4 quadrants.
#define ROWS_PER_TILE 32
#define W_IN 256
#define W_OUT 128

__global__ __launch_bounds__(256) void dwt2d_haar_kernel(
    const float* __restrict__ x, float* __restrict__ out) {
  __shared__ float lds[ROWS_PER_TILE * W_IN];  // 32 KB

  const int t = threadIdx.x;
  // Global flattened row index of this tile's first row: rows are the
  // flattened (b*64+c)*256 + h space. 32 | 256 so tiles never straddle images.
  const unsigned R0 = blockIdx.x * ROWS_PER_TILE;

  // ---- Wave 0 issues the TDM descriptor and waits on TENSORcnt -------------
  if (t < 32) {
    const uint64_t gaddr = (uint64_t)(const void*)x + (uint64_t)R0 * (W_IN * 4);
    // Low 32 bits of the flat shared address == byte offset within LDS.
    const uint32_t lds_addr = (uint32_t)(uintptr_t)(void*)&lds[0];

    // D# group 0 (128b): count=1 | lds_addr | global_addr[56:0] | type=2
    uint32x4 g0;
    g0.x = 1u;                                   // count=1, user descriptor
    g0.y = lds_addr;                             // LDS byte address of tile
    g0.z = (uint32_t)(gaddr & 0xFFFFFFFFu);      // global_addr[31:0]
    g0.w = (uint32_t)((gaddr >> 32) & 0x01FFFFFFu) | 0x80000000u; // [56:32]|type=2

    // D# group 1 (256b): data_size=4B, tensor 256 x 32, tile 256 x 32,
    // dim0 stride = 256 elements, no multicast / pad / iterate / barrier.
    int32x8 g1;
    g1[0] = 0x00020000;                 // workgroup_mask=0, data_size=2 (4B)
    g1[1] = (int)(256u << 16);          // tensor_dim0[15:0] = 256
    g1[2] = (int)((unsigned)ROWS_PER_TILE << 16); // dim0[31:16]=0, tensor_dim1=32
    g1[3] = (int)(256u << 16);          // tensor_dim1[31:16]=0, tile_dim0=256
    g1[4] = ROWS_PER_TILE;              // tile_dim1=32, tile_dim2=0
    g1[5] = W_IN;                       // tensor_dim0_stride[31:0] = 256
    g1[6] = 0;                          // stride[47:32]=0, tensor_dim1_stride lo=0
    g1[7] = 0;

    int32x4 g2 = {0, 0, 0, 0};          // 2D tensor: dims 2/3 unused
    int32x4 g3 = {0, 0, 0, 0};
    int32x8 g4 = {0, 0, 0, 0, 0, 0, 0, 0};  // 5th descriptor group (unused)

    __builtin_amdgcn_tensor_load_to_lds(g0, g1, g2, g3, g4, 0);
    __builtin_amdgcn_s_wait_tensorcnt(0);
  }

  // The compiler does not model the TDM engine's writes to LDS. Escape the
  // LDS pointer into an asm with a memory clobber so subsequent ds_loads are
  // treated as reading data written "by the asm" (i.e. by the DMA engine).
  asm volatile("" ::"r"(&lds[0]) : "memory");

  __syncthreads();  // publish DMA'd tile to all 8 waves

  // ---- Haar butterfly from LDS --------------------------------------------
  const int r = t >> 4;   // 0..15 : output row within tile
  const int g = t & 15;   // 0..15 : group of 8 output columns

  const float* row0 = &lds[(2 * r) * W_IN + 16 * g];
  const float* row1 = row0 + W_IN;

  union { v4f v[4]; float f[16]; } e0, e1;
#pragma unroll
  for (int i = 0; i < 4; ++i) {
    e0.v[i] = ((const v4f*)row0)[i];
    e1.v[i] = ((const v4f*)row1)[i];
  }

  v4f ll[2], lh[2], hl[2], hh[2];
#pragma unroll
  for (int k = 0; k < 8; ++k) {
    const float x00 = e0.f[2 * k + 0];
    const float x01 = e0.f[2 * k + 1];
    const float x10 = e1.f[2 * k + 0];
    const float x11 = e1.f[2 * k + 1];
    const float s0 = x00 + x01, d0 = x00 - x01;
    const float s1 = x10 + x11, d1 = x10 - x11;
    ll[k >> 2][k & 3] = (s0 + s1) * 0.5f;
    lh[k >> 2][k & 3] = (s0 - s1) * 0.5f;
    hl[k >> 2][k & 3] = (d0 + d1) * 0.5f;
    hh[k >> 2][k & 3] = (d0 - d1) * 0.5f;
  }

  // ---- Non-temporal coalesced stores --------------------------------------
  const int img = (int)(R0 >> 8);       // b*64 + c
  const int b   = img >> 6;
  const int c   = img & 63;
  const int i0  = (int)((R0 & 255u) >> 1) + r;        // output row
  const long long qstride = 64LL * W_OUT * W_OUT;     // one quadrant block
  const long long base =
      ((((long long)b * 256 + c) * W_OUT) + i0) * W_OUT + 8 * g;

  float* p = out + base;
  __builtin_nontemporal_store(ll[0], (v4f*)(p));
  __builtin_nontemporal_store(ll[1], (v4f*)(p + 4));
  p += qstride;
  __builtin_nontemporal_store(lh[0], (v4f*)(p));
  __builtin_nontemporal_store(lh[1], (v4f*)(p + 4));
  p += qstride;
  __builtin_nontemporal_store(hl[0], (v4f*)(p));
  __builtin_nontemporal_store(hl[1], (v4f*)(p + 4));
  p += qstride;
  __builtin_nontemporal_store(hh[0], (v4f*)(p));
  __builtin_nontemporal_store(hh[1], (v4f*)(p + 4));
}

extern "C" void kernel_launch(void* const* d_in, const int* in_sizes, int n_in,
                              void* d_out, int out_size, void* d_ws, size_t ws_size,
                              hipStream_t stream) {
  (void)in_sizes; (void)n_in; (void)d_ws; (void)ws_size; (void)out_size;
  const float* x = (const float*)d_in[0];
  float* out = (float*)d_out;

  // 16*64*256 = 262144 flattened input rows, 32 rows per tile -> 8192 blocks.
  const int total_rows = 16 * 64 * 256;
  const int blocks = total_rows / ROWS_PER_TILE;  // 8192
  dwt2d_haar_kernel<<<dim3(blocks), dim3(256), 0, stream>>>(x, out);
}